// ImageWarping_30227979829793
// MI455X (gfx1250) — compile-verified
//
#include <hip/hip_runtime.h>
#include <hip/hip_bf16.h>

// ---------------------------------------------------------------------------
// ImageWarping for MI455X (gfx1250, wave32)
//
// proj = (M @ [x,y,1]) * depth + t   with per-batch M = (K@T)[:3,:3] @ invK33,
//                                           t = (K@T)[:3,3]
// then bilinear border-clamped sample of image at the projected location.
//
// Memory-bound: ~147 MB of traffic -> ~6.3us floor @ 23.3 TB/s. Per-pixel
// compute is trivial fp32 FMA work. The per-batch matrix algebra is done with
// V_WMMA_F32_16X16X4_F32 (one wave per batch).
// ---------------------------------------------------------------------------

typedef float v2f __attribute__((ext_vector_type(2)));
typedef float v8f __attribute__((ext_vector_type(8)));

#define BB 16
#define CC 3
#define HH 512
#define WW 640
#define HW (HH * WW)
#define EPSF 1e-7f

// ---------------------------------------------------------------------------
// Setup: one 32-thread block (one wave) per batch.
//   1) lane0: Gauss-Jordan 4x4 inverse of intrinsics (diag = fx,fy,1,1, no
//      pivoting needed).
//   2) WMMA #1:  P4 = K(4x4) @ pose(4x4)   padded into 16x16x4 f32 WMMA
//   3) WMMA #2:  M  = P[:, :3] @ invK33    padded into 16x16x4 f32 WMMA
//   4) emit 12 floats per batch: M (row-major 3x3) then t (3).
//
// Fragment layouts per ISA sec 7.12.2 (wave32):
//   A 16x4 f32 : lanes 0-15 hold row M=lane, VGPR0=K0, VGPR1=K1;
//                lanes 16-31 hold row M=lane-16, VGPR0=K2, VGPR1=K3.
//   B 4x16 f32 : row-per-VGPR / col-per-lane, K split across lane halves
//                mirroring A: VGPR0 = rows {K0 | K2}, VGPR1 = rows {K1 | K3}.
//   D 16x16 f32: VGPR r, lanes 0-15 -> D[M=r][N=lane] (rows 0..7).
// ---------------------------------------------------------------------------
__global__ void __launch_bounds__(32)
warp_setup_kernel(const float* __restrict__ pose,
                  const float* __restrict__ intr,
                  float* __restrict__ mt) {
  __shared__ float sK[16];    // intrinsics 4x4
  __shared__ float sT[16];    // pose 4x4
  __shared__ float sInv[16];  // inv(intrinsics) 4x4
  __shared__ float sP[16];    // P4 = K @ pose

  const int b = blockIdx.x;
  const int l = threadIdx.x;

  if (l < 16) {
    sK[l] = intr[b * 16 + l];
    sT[l] = pose[b * 16 + l];
  }
  __syncthreads();

  if (l == 0) {
    float a[4][8];
    for (int r = 0; r < 4; ++r)
      for (int c = 0; c < 4; ++c) {
        a[r][c] = sK[r * 4 + c];
        a[r][c + 4] = (r == c) ? 1.0f : 0.0f;
      }
    for (int p = 0; p < 4; ++p) {
      float inv = 1.0f / a[p][p];
      for (int c = 0; c < 8; ++c) a[p][c] *= inv;
      for (int r = 0; r < 4; ++r) {
        if (r == p) continue;
        float f = a[r][p];
        for (int c = 0; c < 8; ++c) a[r][c] -= f * a[p][c];
      }
    }
    for (int r = 0; r < 4; ++r)
      for (int c = 0; c < 4; ++c) sInv[r * 4 + c] = a[r][c + 4];
  }
  __syncthreads();

  const int half = l >> 4;   // which lane half (selects K pair)
  const int lr   = l & 15;   // row (A) / column (B) index
  const int k0 = half * 2;
  const int k1 = half * 2 + 1;

  // ---- WMMA #1: P4 = K @ pose --------------------------------------------
  v2f A1, B1;
  A1.x = (lr < 4) ? sK[lr * 4 + k0] : 0.0f;     // A[row=lr][K=k0]
  A1.y = (lr < 4) ? sK[lr * 4 + k1] : 0.0f;     // A[row=lr][K=k1]
  B1.x = (lr < 4) ? sT[k0 * 4 + lr] : 0.0f;     // B[K=k0][col=lr]
  B1.y = (lr < 4) ? sT[k1 * 4 + lr] : 0.0f;     // B[K=k1][col=lr]
  v8f Cz = {};
  v8f D1 = __builtin_amdgcn_wmma_f32_16x16x4_f32(
      false, A1, false, B1, (short)0, Cz, false, false);

  if (half == 0 && lr < 4) {                    // rows 0..3, cols 0..3 valid
    sP[0 * 4 + lr] = D1[0];
    sP[1 * 4 + lr] = D1[1];
    sP[2 * 4 + lr] = D1[2];
    sP[3 * 4 + lr] = D1[3];
  }
  __syncthreads();

  // ---- WMMA #2: M = P[:, :3] @ invK33 ------------------------------------
  v2f A2, B2;
  A2.x = (lr < 3 && k0 < 3) ? sP[lr * 4 + k0] : 0.0f;
  A2.y = (lr < 3 && k1 < 3) ? sP[lr * 4 + k1] : 0.0f;
  B2.x = (lr < 3 && k0 < 3) ? sInv[k0 * 4 + lr] : 0.0f;
  B2.y = (lr < 3 && k1 < 3) ? sInv[k1 * 4 + lr] : 0.0f;
  v8f D2 = __builtin_amdgcn_wmma_f32_16x16x4_f32(
      false, A2, false, B2, (short)0, Cz, false, false);

  float* o = mt + b * 12;
  if (half == 0 && lr < 3) {     // lane lr holds column lr of M
    o[0 * 3 + lr] = D2[0];       // M[0][lr]
    o[1 * 3 + lr] = D2[1];       // M[1][lr]
    o[2 * 3 + lr] = D2[2];       // M[2][lr]
  }
  if (l == 0) {                  // t = P[:,3]
    o[9]  = sP[0 * 4 + 3];
    o[10] = sP[1 * 4 + 3];
    o[11] = sP[2 * 4 + 3];
  }
}

// ---------------------------------------------------------------------------
// Main warp + bilinear sample kernel. One pixel per thread, batch uniform per
// block (matrix coefficients become scalar loads). Bandwidth-bound.
// ---------------------------------------------------------------------------
__global__ void __launch_bounds__(256)
warp_sample_kernel(const float* __restrict__ img,
                   const float* __restrict__ depth,
                   const float* __restrict__ mt,
                   float* __restrict__ out) {
  constexpr int BLOCKS_PER_BATCH = HW / 256;  // 1280
  const int b = blockIdx.x / BLOCKS_PER_BATCH;
  const int p = (blockIdx.x % BLOCKS_PER_BATCH) * 256 + threadIdx.x;

  const float* __restrict__ mb = mt + b * 12;   // uniform -> s_load
  const float M00 = mb[0], M01 = mb[1], M02 = mb[2];
  const float M10 = mb[3], M11 = mb[4], M12 = mb[5];
  const float M20 = mb[6], M21 = mb[7], M22 = mb[8];
  const float t0 = mb[9], t1 = mb[10], t2 = mb[11];

  const int y = p / WW;
  const int x = p - y * WW;
  const float fx = (float)x;
  const float fy = (float)y;

  const float* __restrict__ base = img + (size_t)b * CC * HW;

  // Warm the cacheline at the identity location: pose is a small perturbation
  // of identity, so the bilinear taps land within a few pixels of (x, y).
  __builtin_prefetch(base + y * WW + x, 0, 1);   // global_prefetch_b8

  const float px = fmaf(M00, fx, fmaf(M01, fy, M02));
  const float py = fmaf(M10, fx, fmaf(M11, fy, M12));
  const float pz = fmaf(M20, fx, fmaf(M21, fy, M22));

  const float d = depth[(size_t)b * HW + p];

  const float X = fmaf(px, d, t0);
  const float Y = fmaf(py, d, t1);
  const float Z = fmaf(pz, d, t2) + EPSF;

  // v_rcp_f32 + 1 Newton step (~full fp32 accuracy)
  float rz = __builtin_amdgcn_rcpf(Z);
  rz = rz * fmaf(-Z, rz, 2.0f);
  const float u = X * rz;
  const float v = Y * rz;

  // unnormalize (align_corners=False) + border clamp:
  // ix = u * W/(W-1) - 0.5, clamped to [0, W-1]
  float ix = fmaf(u, (float)WW / (float)(WW - 1), -0.5f);
  float iy = fmaf(v, (float)HH / (float)(HH - 1), -0.5f);
  ix = fminf(fmaxf(ix, 0.0f), (float)(WW - 1));
  iy = fminf(fmaxf(iy, 0.0f), (float)(HH - 1));

  const float x0f = floorf(ix);
  const float y0f = floorf(iy);
  const float wx = ix - x0f;
  const float wy = iy - y0f;
  const int x0 = (int)x0f;                 // already in [0, W-1]
  const int y0 = (int)y0f;                 // already in [0, H-1]
  const int x1 = (x0 + 1 < WW) ? x0 + 1 : WW - 1;
  const int y1 = (y0 + 1 < HH) ? y0 + 1 : HH - 1;

  const float w11 = wx * wy;
  const float w01 = wx - w11;              // wx*(1-wy)
  const float w10 = wy - w11;              // (1-wx)*wy
  const float w00 = 1.0f - wx - w10;       // (1-wx)*(1-wy)

  const int i00 = y0 * WW + x0;
  const int i01 = y0 * WW + x1;
  const int i10 = y1 * WW + x0;
  const int i11 = y1 * WW + x1;

  float* __restrict__ ob = out + (size_t)b * CC * HW + p;
#pragma unroll
  for (int c = 0; c < CC; ++c) {
    const float* __restrict__ pl = base + c * HW;
    const float r = pl[i00] * w00 + pl[i01] * w01 +
                    pl[i10] * w10 + pl[i11] * w11;
    ob[(size_t)c * HW] = r;
  }
}

// ---------------------------------------------------------------------------
extern "C" void kernel_launch(void* const* d_in, const int* in_sizes, int n_in,
                              void* d_out, int out_size, void* d_ws, size_t ws_size,
                              hipStream_t stream) {
  const float* image = (const float*)d_in[0];   // [B,3,H,W]
  const float* depth = (const float*)d_in[1];   // [B,1,H,W]
  const float* pose  = (const float*)d_in[2];   // [B,4,4]
  const float* intr  = (const float*)d_in[3];   // [B,4,4]
  float* out = (float*)d_out;                   // [B,3,H,W]
  float* mt  = (float*)d_ws;                    // 16 * 12 floats

  warp_setup_kernel<<<BB, 32, 0, stream>>>(pose, intr, mt);

  const int nblocks = (BB * HW) / 256;          // 20480
  warp_sample_kernel<<<nblocks, 256, 0, stream>>>(image, depth, mt, out);
}